// MDDecoder_56788057588116
// MI455X (gfx1250) — compile-verified
//
#include <hip/hip_runtime.h>
#include <hip/hip_bf16.h>
#include <math.h>

// ---------------------------------------------------------------------------
// Problem constants (from reference)
// ---------------------------------------------------------------------------
#define NN      4096
#define DD      128
#define EPSV    1e-7f
#define MINCL   1e-15f
#define MAXSQ   50.0f

#define X_ELEMS      (NN * DD)                 // 524288
#define DIST_ELEMS   (NN * NN)                 // 16777216
#define OUT_LOSS     (X_ELEMS + DIST_ELEMS)    // 17301504
#define OUT_MAXD     (OUT_LOSS + 1)
#define OUT_MAXE     (OUT_LOSS + 2)
#define OUT_MINE     (OUT_LOSS + 3)

// workspace layout (float offsets)
#define WS_NEGPART   0                         // 4096 * 128
#define WS_NEGSUM    (WS_NEGPART + NN * 128)   // 4096
#define WS_BMAXD     (WS_NEGSUM + NN)          // 4096 (64x64 blocks)
#define WS_BMAXE     (WS_BMAXD + 4096)
#define WS_BMINE     (WS_BMAXE + 4096)
#define WS_LOSSP     (WS_BMINE + 4096)         // 65536
#define LOSS_BLOCKS  65536

typedef float v2f __attribute__((ext_vector_type(2)));
typedef float v8f __attribute__((ext_vector_type(8)));

// ---------------------------------------------------------------------------
// Kernel 0: copy x to output (reference returns x unchanged as output[0])
// ---------------------------------------------------------------------------
__global__ __launch_bounds__(256) void init_kernel(const float* __restrict__ x,
                                                   float* __restrict__ out_x) {
    int idx = blockIdx.x * 256 + threadIdx.x;
    if (idx < X_ELEMS) out_x[idx] = x[idx];
}

// ---------------------------------------------------------------------------
// Kernel 1: fused Minkowski GEMM (fp32 WMMA) + hyperbolic-distance epilogue.
// Block = 128 threads (4 waves). Block tile 64x64, wave tile 32x32 (2x2 WMMA).
//   mink[i][j] = sum_k x[i][k] * y[j][k],  y = x with column 0 negated.
// ---------------------------------------------------------------------------
__global__ __launch_bounds__(128) void hyp_gemm_kernel(
    const float* __restrict__ x, const int* __restrict__ adj,
    float* __restrict__ dist_out, float* __restrict__ negpart,
    float* __restrict__ blkMaxD, float* __restrict__ blkMaxE,
    float* __restrict__ blkMinE) {

    const int lane  = threadIdx.x & 31;
    const int w     = threadIdx.x >> 5;      // wave id 0..3
    const int waveM = w >> 1;                // 2x2 wave grid
    const int waveN = w & 1;
    const int half  = lane >> 4;             // 0: lanes 0-15, 1: lanes 16-31
    const int l15   = lane & 15;

    const int rowBase = blockIdx.y * 64 + waveM * 32;
    const int colBase = blockIdx.x * 64 + waveN * 32;

    // A rows this lane supplies (same rows for both halves; halves differ in K)
    const int rA0 = rowBase + l15;
    const int rA1 = rA0 + 16;
    const int cB0 = colBase + l15;
    const int cB1 = cB0 + 16;
    const int koff = half * 2;               // K offset within the 4-wide step

    v8f cc[2][2];
    cc[0][0] = v8f{}; cc[0][1] = v8f{}; cc[1][0] = v8f{}; cc[1][1] = v8f{};

    #pragma unroll 4
    for (int kk = 0; kk < DD; kk += 4) {
        v2f a0 = *(const v2f*)(x + rA0 * DD + kk + koff);
        v2f a1 = *(const v2f*)(x + rA1 * DD + kk + koff);
        v2f b0 = *(const v2f*)(x + cB0 * DD + kk + koff);
        v2f b1 = *(const v2f*)(x + cB1 * DD + kk + koff);
        if (kk == 0 && half == 0) { b0.x = -b0.x; b1.x = -b1.x; }  // Minkowski metric

        cc[0][0] = __builtin_amdgcn_wmma_f32_16x16x4_f32(false, a0, false, b0,
                                                         (short)0, cc[0][0], false, false);
        cc[0][1] = __builtin_amdgcn_wmma_f32_16x16x4_f32(false, a0, false, b1,
                                                         (short)0, cc[0][1], false, false);
        cc[1][0] = __builtin_amdgcn_wmma_f32_16x16x4_f32(false, a1, false, b0,
                                                         (short)0, cc[1][0], false, false);
        cc[1][1] = __builtin_amdgcn_wmma_f32_16x16x4_f32(false, a1, false, b1,
                                                         (short)0, cc[1][1], false, false);
    }

    // ---- fused epilogue ----
    float maxD = -1e30f, maxE = -1e30f, minE = 1e30f;

    #pragma unroll
    for (int mi = 0; mi < 2; ++mi) {
        float rowsum[8];
        #pragma unroll
        for (int r = 0; r < 8; ++r) rowsum[r] = 0.0f;

        #pragma unroll
        for (int ni = 0; ni < 2; ++ni) {
            #pragma unroll
            for (int r = 0; r < 8; ++r) {
                // C layout: VGPR r -> row (r + 8*half), lane l15 -> col
                const int row = rowBase + mi * 16 + r + half * 8;
                const int col = colBase + ni * 16 + l15;
                const int idx = row * NN + col;

                float m     = cc[mi][ni][r];
                float theta = fmaxf(-m, 1.0f + EPSV);
                float ac    = acoshf(theta);
                float dist  = fminf(ac * ac, MAXSQ);
                dist_out[idx] = dist;

                float posf = (float)adj[idx];
                float simi = fmaxf(expf(-dist), MINCL);
                float edge = m * posf;

                maxD = fmaxf(maxD, dist);
                maxE = fmaxf(maxE, edge);
                minE = fminf(minE, edge);
                rowsum[r] += simi * (1.0f - posf);
            }
        }

        // reduce rowsum across the 16 lanes of each half (masks < 16 stay in-half)
        #pragma unroll
        for (int r = 0; r < 8; ++r) {
            float v = rowsum[r];
            for (int mmask = 1; mmask < 16; mmask <<= 1)
                v += __shfl_xor(v, mmask, 32);
            if (l15 == 0) {
                const int row = rowBase + mi * 16 + r + half * 8;
                negpart[row * 128 + blockIdx.x * 2 + waveN] = v;  // unique owner
            }
        }
    }

    // wave reduction of the scalars, then per-block partial
    for (int mmask = 1; mmask < 32; mmask <<= 1) {
        maxD = fmaxf(maxD, __shfl_xor(maxD, mmask, 32));
        maxE = fmaxf(maxE, __shfl_xor(maxE, mmask, 32));
        minE = fminf(minE, __shfl_xor(minE, mmask, 32));
    }
    __shared__ float sD[4], sE[4], sN[4];
    if (lane == 0) { sD[w] = maxD; sE[w] = maxE; sN[w] = minE; }
    __syncthreads();
    if (threadIdx.x == 0) {
        float d = sD[0], e = sE[0], n = sN[0];
        for (int i = 1; i < 4; ++i) {
            d = fmaxf(d, sD[i]); e = fmaxf(e, sE[i]); n = fminf(n, sN[i]);
        }
        const int bid = blockIdx.y * 64 + blockIdx.x;
        blkMaxD[bid] = d; blkMaxE[bid] = e; blkMinE[bid] = n;
    }
}

// ---------------------------------------------------------------------------
// Kernel 2: negsum[row] = sum over 128 column-chunk partials
// ---------------------------------------------------------------------------
__global__ __launch_bounds__(128) void negsum_kernel(const float* __restrict__ negpart,
                                                     float* __restrict__ negsum) {
    int row = blockIdx.x * 128 + threadIdx.x;
    if (row < NN) {
        float s = 0.0f;
        #pragma unroll 8
        for (int c = 0; c < 128; ++c) s += negpart[row * 128 + c];
        negsum[row] = s;
    }
}

// ---------------------------------------------------------------------------
// Kernel 3: reduce 4096 block partials -> three scalar outputs (deterministic)
// ---------------------------------------------------------------------------
__global__ __launch_bounds__(256) void scalar_kernel(const float* __restrict__ bD,
                                                     const float* __restrict__ bE,
                                                     const float* __restrict__ bN,
                                                     float* __restrict__ out) {
    __shared__ float sD[256], sE[256], sN[256];
    float d = -1e30f, e = -1e30f, n = 1e30f;
    for (int i = threadIdx.x; i < 4096; i += 256) {
        d = fmaxf(d, bD[i]); e = fmaxf(e, bE[i]); n = fminf(n, bN[i]);
    }
    sD[threadIdx.x] = d; sE[threadIdx.x] = e; sN[threadIdx.x] = n;
    __syncthreads();
    for (int o = 128; o > 0; o >>= 1) {
        if (threadIdx.x < o) {
            sD[threadIdx.x] = fmaxf(sD[threadIdx.x], sD[threadIdx.x + o]);
            sE[threadIdx.x] = fmaxf(sE[threadIdx.x], sE[threadIdx.x + o]);
            sN[threadIdx.x] = fminf(sN[threadIdx.x], sN[threadIdx.x + o]);
        }
        __syncthreads();
    }
    if (threadIdx.x == 0) { out[1] = sD[0]; out[2] = sE[0]; out[3] = sN[0]; }
}

// ---------------------------------------------------------------------------
// Kernel 4: per-block loss partial sums (edges only), deterministic tree
// ---------------------------------------------------------------------------
__global__ __launch_bounds__(256) void loss_part_kernel(const float* __restrict__ dist,
                                                        const int* __restrict__ adj,
                                                        const float* __restrict__ negsum,
                                                        float* __restrict__ lossPart) {
    const int idx = blockIdx.x * 256 + threadIdx.x;     // < 16777216
    float contrib = 0.0f;
    if (adj[idx] != 0) {
        const int i = idx >> 12;
        float d     = dist[idx];
        float simi  = fmaxf(expf(-d), MINCL);
        float ratio = fmaxf(simi / negsum[i], MINCL);
        contrib     = -logf(ratio);
    }
    __shared__ float s[256];
    s[threadIdx.x] = contrib;
    __syncthreads();
    for (int o = 128; o > 0; o >>= 1) {
        if (threadIdx.x < o) s[threadIdx.x] += s[threadIdx.x + o];
        __syncthreads();
    }
    if (threadIdx.x == 0) lossPart[blockIdx.x] = s[0];
}

// ---------------------------------------------------------------------------
// Kernel 5: final loss reduction (single block, deterministic)
// ---------------------------------------------------------------------------
__global__ __launch_bounds__(256) void loss_final_kernel(const float* __restrict__ lossPart,
                                                         float* __restrict__ lossOut) {
    __shared__ float s[256];
    float acc = 0.0f;
    for (int i = threadIdx.x; i < LOSS_BLOCKS; i += 256) acc += lossPart[i];
    s[threadIdx.x] = acc;
    __syncthreads();
    for (int o = 128; o > 0; o >>= 1) {
        if (threadIdx.x < o) s[threadIdx.x] += s[threadIdx.x + o];
        __syncthreads();
    }
    if (threadIdx.x == 0) *lossOut = s[0];
}

// ---------------------------------------------------------------------------
extern "C" void kernel_launch(void* const* d_in, const int* in_sizes, int n_in,
                              void* d_out, int out_size, void* d_ws, size_t ws_size,
                              hipStream_t stream) {
    const float* x   = (const float*)d_in[0];
    const int*   adj = (const int*)d_in[1];
    float* out       = (float*)d_out;
    float* ws        = (float*)d_ws;

    float* out_x    = out;                 // [0 .. 524287]
    float* out_dist = out + X_ELEMS;       // [524288 .. 17301503]
    float* out_scal = out + OUT_LOSS;      // loss, maxdist, maxedge, minedge

    float* negpart  = ws + WS_NEGPART;
    float* negsum   = ws + WS_NEGSUM;
    float* bMaxD    = ws + WS_BMAXD;
    float* bMaxE    = ws + WS_BMAXE;
    float* bMinE    = ws + WS_BMINE;
    float* lossPart = ws + WS_LOSSP;

    init_kernel<<<(X_ELEMS + 255) / 256, 256, 0, stream>>>(x, out_x);

    dim3 g1(NN / 64, NN / 64);             // 64 x 64 blocks
    hyp_gemm_kernel<<<g1, 128, 0, stream>>>(x, adj, out_dist, negpart,
                                            bMaxD, bMaxE, bMinE);

    negsum_kernel<<<NN / 128, 128, 0, stream>>>(negpart, negsum);

    scalar_kernel<<<1, 256, 0, stream>>>(bMaxD, bMaxE, bMinE, out_scal);

    loss_part_kernel<<<LOSS_BLOCKS, 256, 0, stream>>>(out_dist, adj, negsum, lossPart);

    loss_final_kernel<<<1, 256, 0, stream>>>(lossPart, out_scal);
}